// AGRAN_anchor_88682484728385
// MI455X (gfx1250) — compile-verified
//
#include <hip/hip_runtime.h>
#include <hip/hip_bf16.h>
#include <cstdint>

#define BDIM 16
#define LDIM 200
#define HDIM 128
#define NHD  2
#define DDIM 64
// float(-2**32 + 1) rounds to -4.294967296e9f
#define NEGV -4294967296.0f

typedef float v2f __attribute__((ext_vector_type(2)));
typedef float v8f __attribute__((ext_vector_type(8)));

__device__ __forceinline__ v8f wmma_f32_16x16x4(v2f a, v2f b, v8f c) {
  // 8 args: (neg_a, A, neg_b, B, c_mod, C, reuse_a, reuse_b)
  return __builtin_amdgcn_wmma_f32_16x16x4_f32(false, a, false, b, (short)0, c,
                                               false, false);
}

// ---------------------------------------------------------------------------
// Kernel 1: projections. C[m,n] = sum_c X[m,c] * W[n,c]  (m = b*L+l, n in H)
//   z=0: Qs   = (queries@WqT + bq) * (1/sqrt(D))   (scale folded into Q)
//   z=1: KpK  = keys@WkT + bk + abs_pos_K
//   z=2: Vsum = keys@WvT + bv + abs_pos_V
// Output layout: [b, h, l, d] (head-major), one wave per 16x16 tile.
// ---------------------------------------------------------------------------
__global__ void proj_kernel(const float* __restrict__ queries,
                            const float* __restrict__ keys,
                            const float* __restrict__ Wq, const float* __restrict__ bq,
                            const float* __restrict__ Wk, const float* __restrict__ bk,
                            const float* __restrict__ Wv, const float* __restrict__ bv,
                            const float* __restrict__ pK, const float* __restrict__ pV,
                            float* __restrict__ Qs, float* __restrict__ KpK,
                            float* __restrict__ Vsum) {
  const int t = threadIdx.x;            // 0..31 (one wave32)
  const int mtile = blockIdx.x;         // 0..199 : rows of B*L (3200 = 200*16)
  const int ntile = blockIdx.y;         // 0..7   : cols of H (128 = 8*16)
  const int z = blockIdx.z;             // 0:Q 1:K 2:V

  const float* X    = (z == 0) ? queries : keys;
  const float* W    = (z == 0) ? Wq : (z == 1) ? Wk : Wv;
  const float* bias = (z == 0) ? bq : (z == 1) ? bk : bv;

  const int m = mtile * 16 + (t & 15);  // A row (lane = M)
  const int n = ntile * 16 + (t & 15);  // B col (lane = N)
  const int khalf = 2 * (t >> 4);       // K sub-slot per half-wave

  v8f acc = {0.f, 0.f, 0.f, 0.f, 0.f, 0.f, 0.f, 0.f};
  for (int kb = 0; kb < HDIM; kb += 4) {
    const int kk = kb + khalf;
    float2 av = *(const float2*)(X + (size_t)m * HDIM + kk);
    float2 bw = *(const float2*)(W + (size_t)n * HDIM + kk);
    v2f a; a.x = av.x; a.y = av.y;
    v2f b; b.x = bw.x; b.y = bw.y;
    acc = wmma_f32_16x16x4(a, b, acc);
  }

  const int nn = ntile * 16 + (t & 15);
  const int h = nn >> 6, d = nn & 63;
  for (int r = 0; r < 8; ++r) {
    const int mm = mtile * 16 + r + ((t >= 16) ? 8 : 0);
    const int b_ = mm / LDIM, lq = mm % LDIM;
    const size_t dst = ((size_t)(b_ * NHD + h) * LDIM + lq) * DDIM + d;
    const float v = acc[r];
    if (z == 0) {
      Qs[dst] = (v + bias[nn]) * 0.125f;  // 1/sqrt(64)
    } else if (z == 1) {
      KpK[dst] = v + bias[nn] + pK[(size_t)mm * HDIM + nn];
    } else {
      Vsum[dst] = v + bias[nn] + pV[(size_t)mm * HDIM + nn];
    }
  }
}

// ---------------------------------------------------------------------------
// Kernel 2: base scores S[q,k] = Qs[q,:] . KpK[k,:] per (b,h). WMMA tiles,
// L=200 padded to 13 tiles of 16 with zero-fill / guarded stores.
// ---------------------------------------------------------------------------
__global__ void scores_gemm(const float* __restrict__ Qs,
                            const float* __restrict__ KpK,
                            float* __restrict__ Sg) {
  const int t = threadIdx.x;
  const int qtile = blockIdx.x, ktile = blockIdx.y, bh = blockIdx.z;
  const int q  = qtile * 16 + (t & 15);
  const int kc = ktile * 16 + (t & 15);
  const int khalf = 2 * (t >> 4);
  const bool qin = (q < LDIM), kin = (kc < LDIM);
  const float* Qrow = Qs  + ((size_t)bh * LDIM + (qin ? q : 0)) * DDIM;
  const float* Krow = KpK + ((size_t)bh * LDIM + (kin ? kc : 0)) * DDIM;

  v8f acc = {0.f, 0.f, 0.f, 0.f, 0.f, 0.f, 0.f, 0.f};
  for (int kb = 0; kb < DDIM; kb += 4) {
    const int kk = kb + khalf;
    float2 av = *(const float2*)(Qrow + kk);
    float2 bw = *(const float2*)(Krow + kk);
    v2f a; a.x = qin ? av.x : 0.f; a.y = qin ? av.y : 0.f;
    v2f b; b.x = kin ? bw.x : 0.f; b.y = kin ? bw.y : 0.f;
    acc = wmma_f32_16x16x4(a, b, acc);
  }
  for (int r = 0; r < 8; ++r) {
    const int qq = qtile * 16 + r + ((t >= 16) ? 8 : 0);
    const int kk0 = ktile * 16 + (t & 15);
    if (qq < LDIM && kk0 < LDIM)
      Sg[((size_t)bh * LDIM + qq) * LDIM + kk0] = acc[r];
  }
}

// ---------------------------------------------------------------------------
// Kernel 3: relative-K terms + masking + softmax. One block per (b,q).
// Streams (tmK+dmK)[b,q,k,:] only for k <= q; skips entirely for masked rows.
// ---------------------------------------------------------------------------
__global__ void rel_softmax(const float* __restrict__ Qs,
                            const float* __restrict__ tmK,
                            const float* __restrict__ dmK,
                            const int* __restrict__ time_mask,
                            float* __restrict__ Sg) {
  __shared__ float qv[HDIM];
  __shared__ float srow[NHD][LDIM];
  const int t = threadIdx.x;           // 0..255 (8 waves)
  const int bq = blockIdx.x;           // b*L + q
  const int b_ = bq / LDIM, q = bq % LDIM;
  const bool rowMasked = (time_mask[bq] != 0);

  if (t < HDIM) {
    const int h = t >> 6, d = t & 63;
    qv[t] = Qs[((size_t)(b_ * NHD + h) * LDIM + q) * DDIM + d];
  }
  for (int idx = t; idx < NHD * LDIM; idx += 256)
    srow[idx / LDIM][idx % LDIM] = NEGV;
  __syncthreads();

  if (!rowMasked) {
    const int w = t >> 5, lane = t & 31;
    // half-wave per head: lanes 0-15 -> h=0, lanes 16-31 -> h=1, 4 ch/lane
    const int c4 = ((lane & 15) << 2) + ((lane >> 4) << 6);
    for (int k = w; k <= q; k += 8) {
      const size_t off = ((size_t)bq * LDIM + k) * HDIM + c4;
      float4 a4 = *(const float4*)(tmK + off);
      float4 b4 = *(const float4*)(dmK + off);
      float p = (a4.x + b4.x) * qv[c4]     + (a4.y + b4.y) * qv[c4 + 1]
              + (a4.z + b4.z) * qv[c4 + 2] + (a4.w + b4.w) * qv[c4 + 3];
      p += __shfl_down(p, 8, 16);
      p += __shfl_down(p, 4, 16);
      p += __shfl_down(p, 2, 16);
      p += __shfl_down(p, 1, 16);
      if ((lane & 15) == 0) {
        const int h = lane >> 4;
        srow[h][k] = Sg[((size_t)(b_ * NHD + h) * LDIM + q) * LDIM + k] + p;
      }
    }
  }
  __syncthreads();

  // Softmax: wave 0 -> head 0, wave 1 -> head 1.
  const int w = t >> 5, lane = t & 31;
  if (w < NHD) {
    float m = -3.4e38f;
    for (int k = lane; k < LDIM; k += 32) m = fmaxf(m, srow[w][k]);
    for (int off = 16; off >= 1; off >>= 1) m = fmaxf(m, __shfl_xor(m, off, 32));
    float s = 0.f;
    for (int k = lane; k < LDIM; k += 32) {
      const float e = __expf(srow[w][k] - m);  // exact 0 for masked k
      srow[w][k] = e;
      s += e;
    }
    for (int off = 16; off >= 1; off >>= 1) s += __shfl_xor(s, off, 32);
    const float inv = 1.0f / s;
    for (int k = lane; k < LDIM; k += 32) srow[w][k] *= inv;
  }
  __syncthreads();

  for (int idx = t; idx < NHD * LDIM; idx += 256) {
    const int h = idx / LDIM, k = idx % LDIM;
    Sg[((size_t)(b_ * NHD + h) * LDIM + q) * LDIM + k] = srow[h][k];
  }
}

// ---------------------------------------------------------------------------
// Kernel 4: Obase[q,d] = sum_k A[q,k] * Vsum[k,d] per (b,h). WMMA, K padded
// to 208 with zero fragments.
// ---------------------------------------------------------------------------
__global__ void out_gemm(const float* __restrict__ Ag,
                         const float* __restrict__ Vsum,
                         float* __restrict__ Obase) {
  const int t = threadIdx.x;
  const int qtile = blockIdx.x;   // 13
  const int ntile = blockIdx.y;   // 4 (D = 64)
  const int bh = blockIdx.z;      // 32
  const int q = qtile * 16 + (t & 15);
  const int n = ntile * 16 + (t & 15);
  const int khalf = 2 * (t >> 4);
  const bool qin = (q < LDIM);
  const float* Arow = Ag + ((size_t)bh * LDIM + (qin ? q : 0)) * LDIM;

  v8f acc = {0.f, 0.f, 0.f, 0.f, 0.f, 0.f, 0.f, 0.f};
  for (int kb = 0; kb < 208; kb += 4) {
    const int kk = kb + khalf;
    const bool kin = (kk < LDIM);   // kk even, LDIM even -> pair both in/out
    float2 av = kin ? *(const float2*)(Arow + kk) : make_float2(0.f, 0.f);
    v2f a; a.x = (qin && kin) ? av.x : 0.f; a.y = (qin && kin) ? av.y : 0.f;
    v2f b;
    b.x = kin ? Vsum[((size_t)bh * LDIM + kk) * DDIM + n] : 0.f;
    b.y = kin ? Vsum[((size_t)bh * LDIM + kk + 1) * DDIM + n] : 0.f;
    acc = wmma_f32_16x16x4(a, b, acc);
  }
  for (int r = 0; r < 8; ++r) {
    const int qq = qtile * 16 + r + ((t >= 16) ? 8 : 0);
    if (qq < LDIM)
      Obase[((size_t)bh * LDIM + qq) * DDIM + ntile * 16 + (t & 15)] = acc[r];
  }
}

// ---------------------------------------------------------------------------
// Kernel 5: relative-V terms + final output. One block per (b,q); streams
// (tmV+dmV)[b,q,k,:] for k < kmax (q+1, or L when row masked -> uniform A).
// ---------------------------------------------------------------------------
__global__ void out_rel(const float* __restrict__ Ag,
                        const float* __restrict__ tmV,
                        const float* __restrict__ dmV,
                        const float* __restrict__ Obase,
                        const int* __restrict__ time_mask,
                        float* __restrict__ out) {
  __shared__ float arow[NHD][LDIM];
  __shared__ float4 accbuf[256];
  const int t = threadIdx.x;
  const int bq = blockIdx.x;
  const int b_ = bq / LDIM, q = bq % LDIM;
  const bool rowMasked = (time_mask[bq] != 0);

  for (int idx = t; idx < NHD * LDIM; idx += 256) {
    const int h = idx / LDIM, k = idx % LDIM;
    arow[h][k] = Ag[((size_t)(b_ * NHD + h) * LDIM + q) * LDIM + k];
  }
  __syncthreads();

  const int kmax = rowMasked ? LDIM : (q + 1);
  const int c4 = (t & 31) * 4;     // 4 channels per lane, 32 lanes cover H
  const int kpar = t >> 5;         // 8 k-values in flight
  const int h = c4 >> 6;
  float4 acc = make_float4(0.f, 0.f, 0.f, 0.f);
  for (int k = kpar; k < kmax; k += 8) {
    const size_t off = ((size_t)bq * LDIM + k) * HDIM + c4;
    float4 a4 = *(const float4*)(tmV + off);
    float4 b4 = *(const float4*)(dmV + off);
    const float a = arow[h][k];
    acc.x += a * (a4.x + b4.x);
    acc.y += a * (a4.y + b4.y);
    acc.z += a * (a4.z + b4.z);
    acc.w += a * (a4.w + b4.w);
  }
  accbuf[t] = acc;
  __syncthreads();

  if (t < 32) {
    float4 tot = accbuf[t];
    for (int g = 1; g < 8; ++g) {
      const float4 v = accbuf[t + 32 * g];
      tot.x += v.x; tot.y += v.y; tot.z += v.z; tot.w += v.w;
    }
    const int c0 = t * 4;
    const int hh = c0 >> 6;
    const size_t ob = ((size_t)(b_ * NHD + hh) * LDIM + q) * DDIM + (c0 & 63);
    const size_t oo = (size_t)bq * HDIM + c0;
    out[oo + 0] = tot.x + Obase[ob + 0];
    out[oo + 1] = tot.y + Obase[ob + 1];
    out[oo + 2] = tot.z + Obase[ob + 2];
    out[oo + 3] = tot.w + Obase[ob + 3];
  }
}

// ---------------------------------------------------------------------------
extern "C" void kernel_launch(void* const* d_in, const int* in_sizes, int n_in,
                              void* d_out, int out_size, void* d_ws,
                              size_t ws_size, hipStream_t stream) {
  (void)in_sizes; (void)n_in; (void)out_size; (void)ws_size;
  const float* queries = (const float*)d_in[0];
  const float* keys    = (const float*)d_in[1];
  const float* tmK     = (const float*)d_in[2];
  const float* tmV     = (const float*)d_in[3];
  const float* dmK     = (const float*)d_in[4];
  const float* dmV     = (const float*)d_in[5];
  const float* pK      = (const float*)d_in[6];
  const float* pV      = (const float*)d_in[7];
  const float* Wq      = (const float*)d_in[8];
  const float* bq      = (const float*)d_in[9];
  const float* Wk      = (const float*)d_in[10];
  const float* bk      = (const float*)d_in[11];
  const float* Wv      = (const float*)d_in[12];
  const float* bv      = (const float*)d_in[13];
  const int* time_mask = (const int*)d_in[14];
  // d_in[15] attn_mask is the causal mask: computed analytically (k > q).

  float* ws    = (float*)d_ws;
  float* Qs    = ws;                 // B*NH*L*D = 409600 floats
  float* KpK   = ws + 409600;        // 409600
  float* Vsum  = ws + 819200;        // 409600
  float* Sg    = ws + 1228800;       // B*NH*L*L = 1280000 (scores, then A)
  float* Obase = ws + 3788800;       // 409600
  float* outp  = (float*)d_out;

  proj_kernel<<<dim3(200, 8, 3), 32, 0, stream>>>(
      queries, keys, Wq, bq, Wk, bk, Wv, bv, pK, pV, Qs, KpK, Vsum);
  scores_gemm<<<dim3(13, 13, 32), 32, 0, stream>>>(Qs, KpK, Sg);
  rel_softmax<<<dim3(BDIM * LDIM), 256, 0, stream>>>(Qs, tmK, dmK, time_mask, Sg);
  out_gemm<<<dim3(13, 4, 32), 32, 0, stream>>>(Sg, Vsum, Obase);
  out_rel<<<dim3(BDIM * LDIM), 256, 0, stream>>>(Sg, tmV, dmV, Obase, time_mask, outp);
}